// PythiaAttention_32684701122812
// MI455X (gfx1250) — compile-verified
//
#include <hip/hip_runtime.h>
#include <hip/hip_bf16.h>
#include <math.h>
#include <stdint.h>

#define HIDDEN   2048
#define NHEADS   16
#define HDIM     128
#define SEQ      2048
#define BATCH    2
#define QKV_N    (3 * HIDDEN)

typedef float v2f __attribute__((ext_vector_type(2)));
typedef float v8f __attribute__((ext_vector_type(8)));

#if __has_builtin(__builtin_amdgcn_global_load_async_to_lds_b128) && \
    __has_builtin(__builtin_amdgcn_s_wait_asynccnt)
#define USE_ASYNC 1
#else
#define USE_ASYNC 0
#endif

#if USE_ASYNC
// builtin prototype (from hipcc diagnostic): arg0 = AS(1) int4* (global src),
// arg1 = AS(3) int4* (LDS dst), then imm offset, imm cpol.
typedef int v4i_gcc __attribute__((vector_size(16)));
typedef __attribute__((address_space(1))) v4i_gcc* as1_v4i_p;
typedef __attribute__((address_space(3))) v4i_gcc* as3_v4i_p;
#define ASYNC_WAIT(n) __builtin_amdgcn_s_wait_asynccnt(n)
#else
#define ASYNC_WAIT(n)
#endif

// 16-byte HBM -> LDS copy: async on CDNA5 (ASYNCcnt), sync float4 fallback.
static __device__ __forceinline__ void async_ld16(float* ldst, const float* gsrc) {
#if USE_ASYNC
    __builtin_amdgcn_global_load_async_to_lds_b128(
        (as1_v4i_p)(uintptr_t)gsrc,
        (as3_v4i_p)(unsigned)(uintptr_t)ldst,
        /*imm offset=*/0, /*cpol=*/0);
#else
    *(float4*)ldst = *(const float4*)gsrc;
#endif
}

static __device__ __forceinline__ v8f vzero8() {
    v8f z;
#pragma unroll
    for (int j = 0; j < 8; ++j) z[j] = 0.0f;
    return z;
}

// Exact fp32 WMMA: D(16x16) = A(16x4) * B(4x16) + C
static __device__ __forceinline__ v8f wmma4(v2f a, v2f b, v8f c) {
    return __builtin_amdgcn_wmma_f32_16x16x4_f32(
        /*neg_a=*/false, a, /*neg_b=*/false, b,
        /*c_mod=*/(short)0, c, /*reuse_a=*/false, /*reuse_b=*/false);
}

// ---------------------------------------------------------------------------
// Generic fp32 WMMA GEMM: out = A[MxK] @ B[KxN] + bias[N]
// Block tile 128x128, 256 threads = 8 waves (4x2), wave tile 32x64.
// K-loop double-buffered via async HBM->LDS copies (ASYNCcnt pipelining).
// mode 0: plain row-major out[M][N]
// mode 1: QKV scatter -> out laid out as [B*H][S][384] (q|k|v per head)
// ---------------------------------------------------------------------------
__global__ __launch_bounds__(256) void gemm_wmma_f32(
    const float* __restrict__ A, const float* __restrict__ B,
    const float* __restrict__ bias, float* __restrict__ out,
    int M, int N, int K, int mode)
{
    __shared__ __align__(16) float Alds[2][128][20];   // [m][k], 80B rows (16B aligned)
    __shared__ __align__(16) float Blds[2][16][132];   // [k][n], 528B rows (16B aligned)

    const int tid  = threadIdx.x;
    const int lane = tid & 31;
    const int hi   = lane >> 4;      // 0/1 half of wave
    const int ln   = lane & 15;
    const int wid  = tid >> 5;       // 0..7
    const int wm   = wid >> 1;       // 0..3 (wave row)
    const int wn   = wid & 1;        // 0..1 (wave col)
    const int row0 = blockIdx.y * 128;
    const int col0 = blockIdx.x * 128;

    v8f acc[2][4];
#pragma unroll
    for (int mt = 0; mt < 2; ++mt)
#pragma unroll
        for (int nt = 0; nt < 4; ++nt) acc[mt][nt] = vzero8();

    // per chunk: A = 512 float4, B = 512 float4 -> exactly 4 async ops/thread
    auto issue_chunk = [&](int buf, int c) {
        const int k0 = c << 4;
#pragma unroll
        for (int it = 0; it < 2; ++it) {
            const int e = tid + it * 256;
            {   // A tile 128x16, k-contiguous 16B pieces
                const int r = e >> 2, c4 = e & 3;
                async_ld16(&Alds[buf][r][c4 * 4],
                           A + (size_t)(row0 + r) * K + k0 + c4 * 4);
            }
            {   // B tile 16x128, n-contiguous 16B pieces
                const int kk = e >> 5, c4 = e & 31;
                async_ld16(&Blds[buf][kk][c4 * 4],
                           B + (size_t)(k0 + kk) * N + col0 + c4 * 4);
            }
        }
    };

    const int nch = K >> 4;
    issue_chunk(0, 0);                       // prologue: chunk 0 in flight

    for (int c = 0; c < nch; ++c) {
        const int p = c & 1;
        if (c + 1 < nch) {
            issue_chunk((c + 1) & 1, c + 1); // overlap next tile's HBM traffic
            ASYNC_WAIT(4);                   // keep only the newest group in flight
        } else {
            ASYNC_WAIT(0);
        }
        __syncthreads();

#pragma unroll
        for (int ks = 0; ks < 4; ++ks) {
            const int kb = ks * 4 + 2 * hi;  // this lane's K pair base
            v2f af[2], bf[4];
#pragma unroll
            for (int mt = 0; mt < 2; ++mt)
                af[mt] = *(const v2f*)&Alds[p][wm * 32 + mt * 16 + ln][kb];
#pragma unroll
            for (int nt = 0; nt < 4; ++nt) {
                const int n = wn * 64 + nt * 16 + ln;
                bf[nt].x = Blds[p][kb][n];
                bf[nt].y = Blds[p][kb + 1][n];
            }
#pragma unroll
            for (int mt = 0; mt < 2; ++mt)
#pragma unroll
                for (int nt = 0; nt < 4; ++nt)
                    acc[mt][nt] = wmma4(af[mt], bf[nt], acc[mt][nt]);
        }
        __syncthreads();
    }

    // epilogue: bias + store (C layout: row = i + 8*hi, col = ln)
#pragma unroll
    for (int mt = 0; mt < 2; ++mt) {
#pragma unroll
        for (int nt = 0; nt < 4; ++nt) {
            const int n  = col0 + wn * 64 + nt * 16 + ln;
            const float bv = bias[n];
#pragma unroll
            for (int i = 0; i < 8; ++i) {
                const int m = row0 + wm * 32 + mt * 16 + i + 8 * hi;
                const float v = acc[mt][nt][i] + bv;
                if (mode == 0) {
                    out[(size_t)m * N + n] = v;
                } else {
                    const int b = m >> 11, s = m & 2047;      // token -> (b,s)
                    const int h = n / 384, r = n - h * 384;   // col -> (head, slot)
                    out[((size_t)(b * NHEADS + h) * SEQ + s) * 384 + r] = v;
                }
            }
        }
    }
}

// ---------------------------------------------------------------------------
// RoPE on first 32 dims of q and k, in place on [B*H][S][384] layout.
// One thread per (bh, s, i) pair, i in 0..15 handles dims (i, i+16).
// ---------------------------------------------------------------------------
__global__ __launch_bounds__(256) void rope_kernel(float* __restrict__ qkvh)
{
    const int idx = blockIdx.x * blockDim.x + threadIdx.x;
    const int i  = idx & 15;
    const int s  = (idx >> 4) & 2047;
    const int hb = idx >> 15;                 // b*16 + h, 0..31
    if (hb >= BATCH * NHEADS) return;
    const size_t base = ((size_t)hb * SEQ + s) * 384;

    // inv_freq[i] = 10000^(-2i/32) = exp(-i * ln(10000)/16)
    const float freq = __expf(-(float)i * (9.210340371976184f / 16.0f));
    const float ang  = (float)s * freq;
    const float c  = __cosf(ang);
    const float sn = __sinf(ang);

    float q0 = qkvh[base + i], q1 = qkvh[base + i + 16];
    qkvh[base + i]        = q0 * c - q1 * sn;
    qkvh[base + i + 16]   = q1 * c + q0 * sn;
    float k0 = qkvh[base + 128 + i], k1 = qkvh[base + 128 + i + 16];
    qkvh[base + 128 + i]      = k0 * c - k1 * sn;
    qkvh[base + 128 + i + 16] = k1 * c + k0 * sn;
}

// ---------------------------------------------------------------------------
// Flash attention, fp32 WMMA. Block = 128 threads (4 waves), 64 q-rows/block.
// Wave w owns q rows [qt*64 + w*16, +16). Streams 32-key K/V tiles via
// async HBM->LDS copies.
// ---------------------------------------------------------------------------
__global__ __launch_bounds__(128) void flash_attn(
    const float* __restrict__ qkvh, float* __restrict__ outp)
{
    __shared__ __align__(16) float Klds[32][132];     // [key][d]
    __shared__ __align__(16) float Vlds[32][132];     // [key][d]
    __shared__ __align__(16) float Plds[4][16][36];   // per-wave P transpose

    const int tid  = threadIdx.x;
    const int lane = tid & 31;
    const int w    = tid >> 5;      // wave 0..3
    const int hi   = lane >> 4;
    const int ln   = lane & 15;
    const int qt   = blockIdx.x;    // q tile (64 rows)
    const int h    = blockIdx.y;
    const int b    = blockIdx.z;
    const int bh   = b * NHEADS + h;
    const float scale = 0.08838834764831845f;   // 1/sqrt(128)

    const float* qbase = qkvh + (size_t)bh * SEQ * 384;

    // Q fragments for this wave (A layout: m = ln, K = 4*kk + 2*hi + {0,1})
    const int qrow_frag = qt * 64 + w * 16 + ln;
    const float* qp = qbase + (size_t)qrow_frag * 384;
    v2f qf[32];
#pragma unroll
    for (int kk = 0; kk < 32; ++kk) {
        v2f t = *(const v2f*)(qp + kk * 4 + 2 * hi);
        t.x *= scale; t.y *= scale;
        qf[kk] = t;
    }

    v8f O[8];
#pragma unroll
    for (int nt2 = 0; nt2 < 8; ++nt2) O[nt2] = vzero8();
    float Mrun[8], Lrun[8];
#pragma unroll
    for (int i = 0; i < 8; ++i) { Mrun[i] = -INFINITY; Lrun[i] = 0.0f; }

    const int jt_end = 2 * qt + 1;               // causal: keys <= qt*64+63
    for (int jt = 0; jt <= jt_end; ++jt) {
        // async K/V tile load (32 keys x 128 dims, 16B pieces)
        const float* kp = qbase + (size_t)(jt * 32) * 384 + 128;
        const float* vp = kp + 128;
        for (int idx = tid; idx < 32 * 32; idx += 128) {
            const int r = idx >> 5, c4 = idx & 31;
            async_ld16(&Klds[r][c4 * 4], kp + (size_t)r * 384 + c4 * 4);
            async_ld16(&Vlds[r][c4 * 4], vp + (size_t)r * 384 + c4 * 4);
        }
        ASYNC_WAIT(0);
        __syncthreads();

        const int kb0 = jt * 32;
        if (kb0 <= qt * 64 + w * 16 + 15) {      // wave has at least one valid key
            // S = q @ k^T  (2 column tiles of 16 keys)
            v8f sacc[2];
            sacc[0] = vzero8(); sacc[1] = vzero8();
#pragma unroll
            for (int kk = 0; kk < 32; ++kk) {
                const int kq = kk * 4 + 2 * hi;
#pragma unroll
                for (int nt = 0; nt < 2; ++nt) {
                    v2f bf = *(const v2f*)&Klds[nt * 16 + ln][kq];
                    sacc[nt] = wmma4(qf[kk], bf, sacc[nt]);
                }
            }
            // causal mask on (partially) diagonal tiles
            if (kb0 + 31 > qt * 64 + w * 16) {
#pragma unroll
                for (int nt = 0; nt < 2; ++nt) {
                    const int key = kb0 + nt * 16 + ln;
#pragma unroll
                    for (int i = 0; i < 8; ++i) {
                        const int qrow = qt * 64 + w * 16 + i + 8 * hi;
                        if (key > qrow) sacc[nt][i] = -INFINITY;
                    }
                }
            }
            // online softmax (rows live across 16-lane groups)
#pragma unroll
            for (int i = 0; i < 8; ++i) {
                float rm = fmaxf(sacc[0][i], sacc[1][i]);
                rm = fmaxf(rm, __shfl_xor(rm, 1, 32));
                rm = fmaxf(rm, __shfl_xor(rm, 2, 32));
                rm = fmaxf(rm, __shfl_xor(rm, 4, 32));
                rm = fmaxf(rm, __shfl_xor(rm, 8, 32));
                const float mnew  = fmaxf(Mrun[i], rm);
                const float alpha = __expf(Mrun[i] - mnew);
                const float p0 = __expf(sacc[0][i] - mnew);
                const float p1 = __expf(sacc[1][i] - mnew);
                float rs = p0 + p1;
                rs += __shfl_xor(rs, 1, 32);
                rs += __shfl_xor(rs, 2, 32);
                rs += __shfl_xor(rs, 4, 32);
                rs += __shfl_xor(rs, 8, 32);
                Lrun[i] = Lrun[i] * alpha + rs;
                Mrun[i] = mnew;
                Plds[w][i + 8 * hi][ln]      = p0;   // transpose P via LDS
                Plds[w][i + 8 * hi][16 + ln] = p1;
#pragma unroll
                for (int nt2 = 0; nt2 < 8; ++nt2) O[nt2][i] *= alpha;
            }
            // O += P @ V  (A-frag from Plds, B-frag from Vlds)
            v2f pf[8];
#pragma unroll
            for (int kk2 = 0; kk2 < 8; ++kk2)
                pf[kk2] = *(const v2f*)&Plds[w][ln][kk2 * 4 + 2 * hi];
#pragma unroll
            for (int kk2 = 0; kk2 < 8; ++kk2) {
                const int kv = kk2 * 4 + 2 * hi;
#pragma unroll
                for (int nt2 = 0; nt2 < 8; ++nt2) {
                    v2f vf;
                    vf.x = Vlds[kv][nt2 * 16 + ln];
                    vf.y = Vlds[kv + 1][nt2 * 16 + ln];
                    O[nt2] = wmma4(pf[kk2], vf, O[nt2]);
                }
            }
        }
        __syncthreads();
    }

    // epilogue: O / l -> attn output, token-major [B*S][H*D]
#pragma unroll
    for (int nt2 = 0; nt2 < 8; ++nt2) {
        const int d = nt2 * 16 + ln;
#pragma unroll
        for (int i = 0; i < 8; ++i) {
            const int s = qt * 64 + w * 16 + i + 8 * hi;
            outp[((size_t)(b * SEQ + s)) * HIDDEN + h * HDIM + d] = O[nt2][i] / Lrun[i];
        }
    }
}

// ---------------------------------------------------------------------------
extern "C" void kernel_launch(void* const* d_in, const int* in_sizes, int n_in,
                              void* d_out, int out_size, void* d_ws, size_t ws_size,
                              hipStream_t stream)
{
    const float* X  = (const float*)d_in[0];   // hidden_states [2,2048,2048]
    const float* Wq = (const float*)d_in[1];   // W_qkv [2048,6144]
    const float* bq = (const float*)d_in[2];   // b_qkv [6144]
    const float* Wd = (const float*)d_in[3];   // W_dense [2048,2048]
    const float* bd = (const float*)d_in[4];   // b_dense [2048]
    float* out = (float*)d_out;

    float* qkvh = (float*)d_ws;                                  // [B*H][S][384]
    float* attn = qkvh + (size_t)BATCH * NHEADS * SEQ * 384;     // [B*S][HIDDEN]

    // 1) fused QKV projection + bias, scattered to head-major layout
    dim3 g1(QKV_N / 128, (BATCH * SEQ) / 128);
    gemm_wmma_f32<<<g1, 256, 0, stream>>>(X, Wq, bq, qkvh,
                                          BATCH * SEQ, QKV_N, HIDDEN, 1);

    // 2) RoPE on q,k rotary dims (in place)
    rope_kernel<<<(BATCH * NHEADS * SEQ * 16) / 256, 256, 0, stream>>>(qkvh);

    // 3) causal flash attention
    dim3 g3(SEQ / 64, NHEADS, BATCH);
    flash_attn<<<g3, 128, 0, stream>>>(qkvh, attn);

    // 4) dense output projection + bias
    dim3 g4(HIDDEN / 128, (BATCH * SEQ) / 128);
    gemm_wmma_f32<<<g4, 256, 0, stream>>>(attn, Wd, bd, out,
                                          BATCH * SEQ, HIDDEN, HIDDEN, 0);
}